// RoiPoolingConv_13013750907317
// MI455X (gfx1250) — compile-verified
//
#include <hip/hip_runtime.h>
#include <stdint.h>

#define POOL     7
#define NUM_ROIS 300
#define IMG_H    200
#define IMG_W    200
#define IMG_C    512

#define WAVES_PER_BLOCK 4
#define TILE_FLOATS     2048   // 2 rows x 2 pixels x 512 ch
// LDS: 4 waves * 2 buffers * 2048 floats * 4B = 64 KB

typedef uint32_t u32;
typedef uint64_t u64;
typedef unsigned int uint32x4 __attribute__((ext_vector_type(4)));
typedef int          int32x4  __attribute__((ext_vector_type(4)));
typedef int          int32x8  __attribute__((ext_vector_type(8)));
typedef float        fx4      __attribute__((ext_vector_type(4)));

#if __has_include(<hip/amd_detail/amd_gfx1250_TDM.h>)
#define TDM_SIX_ARGS 1
#endif

__device__ __forceinline__ int imin(int a, int b) { return a < b ? a : b; }
__device__ __forceinline__ int imax(int a, int b) { return a > b ? a : b; }
__device__ __forceinline__ u32 rfl(u32 v) {
    return (u32)__builtin_amdgcn_readfirstlane((int)v);
}

// Issue an async TDM load of a (rows x tile_w_elems) fp32 tile (row stride =
// IMG_W*IMG_C elements) from global `gaddr` into LDS byte address `lds_addr`.
// Tracked by TENSORcnt.
__device__ __forceinline__ void tdm_load_tile(u32 lds_addr, u64 gaddr,
                                              u32 tile_w_elems, u32 rows) {
    const u32 ga_lo = (u32)(gaddr & 0xFFFFFFFFu);
    const u32 ga_hi = (u32)((gaddr >> 32) & 0x01FFFFFFu);

    uint32x4 g0;
    g0.x = rfl(1u);                         // count=1, user D#, no gather
    g0.y = rfl(lds_addr);                   // lds_addr (bytes)
    g0.z = rfl(ga_lo);                      // global_addr[31:0]
    g0.w = rfl(ga_hi | (2u << 30));         // global_addr[56:32] | type=2 (image)

    const u32 tdim0 = tile_w_elems;         // tensor dim0 == tile width (kept in-bounds)
    const u32 tdim1 = rows;
    const u64 stride0 = (u64)IMG_W * IMG_C; // elements between tensor rows

    int32x8 g1;
    g1[0] = (int)rfl(2u << 16);                                      // data_size = 4B
    g1[1] = (int)rfl((tdim0 & 0xFFFFu) << 16);                       // tensor_dim0 lo16
    g1[2] = (int)rfl((tdim0 >> 16) | ((tdim1 & 0xFFFFu) << 16));     // dim0 hi | dim1 lo
    g1[3] = (int)rfl((tdim1 >> 16) | ((tile_w_elems & 0xFFFFu) << 16)); // dim1 hi | tile_dim0
    g1[4] = (int)rfl(rows & 0xFFFFu);                                // tile_dim1 (tile_dim2=0)
    g1[5] = (int)rfl((u32)(stride0 & 0xFFFFFFFFu));                  // dim0_stride[31:0]
    g1[6] = (int)rfl((u32)(stride0 >> 32));                          // dim0_stride[47:32]
    g1[7] = 0;                                                       // dim1_stride hi (2D: unused)

    int32x4 z4 = {0, 0, 0, 0};
#if defined(TDM_SIX_ARGS)
    int32x8 z8 = {0, 0, 0, 0, 0, 0, 0, 0};
    __builtin_amdgcn_tensor_load_to_lds(g0, g1, z4, z4, z8, 0);
#else
    __builtin_amdgcn_tensor_load_to_lds(g0, g1, z4, z4, 0);
#endif
}

__global__ __launch_bounds__(WAVES_PER_BLOCK * 32)
void roi_pool_tdm(const float* __restrict__ img, const float* __restrict__ rois,
                  float* __restrict__ out) {
    __shared__ float sm[WAVES_PER_BLOCK * 2 * TILE_FLOATS];

    const int lane = threadIdx.x & 31;
    const int wv   = threadIdx.x >> 5;
    const int task = blockIdx.x * WAVES_PER_BLOCK + wv;   // one (roi, py) per wave
    if (task >= NUM_ROIS * POOL) return;
    const int roi = task / POOL;
    const int py  = task - roi * POOL;

    // ---- decode ROI (wave-uniform; mirrors reference exactly) ----
    const int r0 = (int)rois[roi * 4 + 0];
    const int r1 = (int)rois[roi * 4 + 1];
    const int r2 = (int)rois[roi * 4 + 2];
    const int r3 = (int)rois[roi * 4 + 3];
    const int x0 = imin(imax(r0, 0), IMG_W - 1);
    const int y0 = imin(imax(r1, 0), IMG_H - 1);
    const int w  = imin(imax(r2, 1), IMG_W - x0);
    const int h  = imin(imax(r3, 1), IMG_H - y0);

    const float ys = (float)py * ((float)h / (float)POOL);
    int iy0        = (int)floorf(ys);
    const float fy = ys - (float)iy0;
    iy0            = imin(iy0, h - 1);
    const int iy1  = imin(iy0 + 1, h - 1);
    const int ay0  = y0 + iy0;
    const int dy   = (y0 + iy1) - ay0;                 // 0 or 1
    const u32 rows = (ay0 + 1 < IMG_H) ? 2u : 1u;

    float* mySm   = &sm[wv * 2 * TILE_FLOATS];
    const u32 lds0 = (u32)(uintptr_t)(void*)mySm;      // flat addr[31:0] == LDS byte offset
    const float xscale = (float)w / (float)POOL;

    // geometry for a given px (wave-uniform)
    auto geom = [&](int px, int& ax0, int& dx, float& fx, int& spanx) {
        const float xs = (float)px * xscale;
        int ix0 = (int)floorf(xs);
        fx      = xs - (float)ix0;
        ix0     = imin(ix0, w - 1);
        const int ix1 = imin(ix0 + 1, w - 1);
        ax0   = x0 + ix0;
        dx    = ix1 - ix0;                             // 0 or 1
        spanx = (ax0 + 1 < IMG_W) ? 2 : 1;
    };

    auto issue = [&](int px, int parity) {
        int ax0, dx, spanx; float fx;
        geom(px, ax0, dx, fx, spanx);
        const u64 gaddr = (u64)(uintptr_t)img +
                          4ull * (((u64)ay0 * IMG_W + (u64)ax0) * IMG_C);
        tdm_load_tile(lds0 + (u32)parity * TILE_FLOATS * 4u, gaddr,
                      (u32)spanx * IMG_C, rows);
    };

    // ---- software pipeline: TDM double-buffer over px ----
    issue(0, 0);
    for (int px = 0; px < POOL; ++px) {
        const int p = px & 1;
        if (px + 1 < POOL) {
            issue(px + 1, (px + 1) & 1);
            __builtin_amdgcn_s_wait_tensorcnt(1);      // tile for `px` has landed
        } else {
            __builtin_amdgcn_s_wait_tensorcnt(0);
        }

        int ax0, dx, spanx; float fx;
        geom(px, ax0, dx, fx, spanx);
        const float w00 = (1.0f - fy) * (1.0f - fx);
        const float w01 = (1.0f - fy) * fx;
        const float w10 = fy * (1.0f - fx);
        const float w11 = fy * fx;

        const float* buf = mySm + p * TILE_FLOATS;
        const int rowS   = spanx * IMG_C;              // LDS row stride (floats)
        float* o = out + ((size_t)(roi * POOL + py) * POOL + px) * IMG_C;

        #pragma unroll
        for (int cc = 0; cc < IMG_C; cc += 128) {
            const int c = cc + lane * 4;
            const fx4 v00 = *(const fx4*)&buf[c];
            const fx4 v01 = *(const fx4*)&buf[dx * IMG_C + c];
            const fx4 v10 = *(const fx4*)&buf[dy * rowS + c];
            const fx4 v11 = *(const fx4*)&buf[dy * rowS + dx * IMG_C + c];
            fx4 r = w00 * v00 + w01 * v01 + w10 * v10 + w11 * v11;
            __builtin_nontemporal_store(r, (fx4*)(o + c));
        }
    }
}

extern "C" void kernel_launch(void* const* d_in, const int* in_sizes, int n_in,
                              void* d_out, int out_size, void* d_ws, size_t ws_size,
                              hipStream_t stream) {
    const float* img  = (const float*)d_in[0];
    const float* rois = (const float*)d_in[1];
    float* out        = (float*)d_out;

    const int tasks = NUM_ROIS * POOL;                         // 2100 waves
    dim3 grid((tasks + WAVES_PER_BLOCK - 1) / WAVES_PER_BLOCK);
    dim3 block(WAVES_PER_BLOCK * 32);
    roi_pool_tdm<<<grid, block, 0, stream>>>(img, rois, out);
}